// FMMPreWork_9904194585100
// MI455X (gfx1250) — compile-verified
//
#include <hip/hip_runtime.h>
#include <math.h>

// ---------------------------------------------------------------------------
// FMM pipeline for MI455X (gfx1250, wave32).
// All GEMM-shaped work (conv1x1, DFT/iDFT stages of rfft2/irfft2) runs through
// a generic V_WMMA_F32_16X16X4_F32 tile kernel (fp32 end-to-end: the workload
// is HBM/L2-bound at ~500MB traffic vs 23.3 TB/s and ~192MB L2, so f32 WMMA
// keeps full reference precision at zero cost). Waves compute 16xNT*16 tiles
// (NT=4 where N%64==0) so one A fragment feeds 4 WMMAs per K-step.
// ---------------------------------------------------------------------------

typedef float v2f __attribute__((ext_vector_type(2)));
typedef float v8f __attribute__((ext_vector_type(8)));

#define GF_ACCUM 1
#define GF_GELU  2

static constexpr int  B_   = 8;
static constexpr int  C_   = 48;
static constexpr int  H_   = 128;
static constexpr int  W_   = 128;
static constexpr long HW_  = (long)H_ * W_;          // 16384
static constexpr long NCf  = (long)B_ * C_ * HW_;    // 6291456 floats per NCHW tensor
static constexpr int  WF_  = 80;                     // 65 rfft bins padded to 80
static constexpr long P80  = (long)H_ * WF_;         // 10240 pixels per freq plane
static constexpr long PLANEf = (long)B_ * C_ * P80;  // 3932160
static constexpr long BIGM = 1L << 30;

// workspace offsets (floats)
static constexpr long o_low = 0;
static constexpr long o_high = 1 * NCf;
static constexpr long o_hf  = 2 * NCf;
static constexpr long o_lfb = 3 * NCf;   // fmgm_low output
static constexpr long o_ho  = 4 * NCf;   // also plane p1 (R1r / zr)
static constexpr long o_lo  = 5 * NCf;   // also plane p2 (R1i / zi)
static constexpr long o_t1  = 6 * NCf;   // 2*NC: kv_pre / frq
static constexpr long o_t2  = 8 * NCf;   // 2*NC: kv / gbuf
static constexpr long o_t3  = 10 * NCf;  // q_pre / outattn / plane p3 (Z2r)
static constexpr long o_t4  = 11 * NCf;  // q / plane p4 (Z2i)
static constexpr long o_tab = 12 * NCf;
static constexpr long o_Tcos  = o_tab;            // [128][80]
static constexpr long o_Tnsin = o_Tcos  + 10240;  // [128][80]
static constexpr long o_Cos2  = o_Tnsin + 10240;  // [128][128]
static constexpr long o_Sin2  = o_Cos2  + 16384;  // [128][128]
static constexpr long o_Wc    = o_Sin2  + 16384;  // [80][128]
static constexpr long o_Ws    = o_Wc    + 10240;  // [80][128]
static constexpr long o_sm    = o_Ws    + 10240;
static constexpr long o_ap    = o_sm;             // 384
static constexpr long o_lfraw = o_sm + 512;       // 576
static constexpr long o_lfw   = o_sm + 1536;      // 576
static constexpr long o_qn    = o_sm + 2560;      // 384
static constexpr long o_kn    = o_sm + 3072;      // 384
static constexpr long o_dots  = o_sm + 3584;      // 4608
static constexpr long o_attn  = o_sm + 8192;      // 4608

__device__ inline float dev_gelu(float x) {
  return 0.5f * x * (1.0f + erff(x * 0.70710678118654752f));
}
__device__ inline int refl(int i, int n) {
  return i < 0 ? -i : (i >= n ? 2 * n - 2 - i : i);
}

// ---------------------------------------------------------------------------
// Generic fp32 WMMA GEMM: D[m][n] (+)= alpha * sum_k A[m][k]*B[k][n]
//   [+bias[m]] [gelu] [+resid].  Batch offsets: (t%mod)*s1 + (t/mod)*s2.
// Grid: x = N/(16*NT), y = M/16, z = batch. One wave per 16 x (16*NT) tile:
// the A fragment is loaded once per K-step and feeds NT WMMAs.
// A layout (16x4):  lane = (k&2 ? 16:0)+m, vgpr = k&1
// B layout (4x16):  lane = (k&2 ? 16:0)+n, vgpr = k&1
// C/D (16x16):      vgpr i holds rows i (lanes 0-15) and i+8 (lanes 16-31)
// ---------------------------------------------------------------------------
template <int NT>
__global__ void gemm_wmma_f32(
    const float* __restrict__ A, int lda, long am, long as1, long as2,
    const float* __restrict__ Bm, int ldb, long bm, long bs1, long bs2,
    float* __restrict__ D, int ldd, long dm, long ds1, long ds2,
    int K, float alpha, const float* __restrict__ bias,
    const float* __restrict__ resid, int flags) {
  const int lane = threadIdx.x;
  const int n0 = blockIdx.x * (16 * NT);
  const int m0 = blockIdx.y * 16;
  const long t = blockIdx.z;
  const float* Ab = A + (t % am) * as1 + (t / am) * as2 + (long)m0 * lda;
  const float* Bb = Bm + (t % bm) * bs1 + (t / bm) * bs2 + n0;
  const long offD = (t % dm) * ds1 + (t / dm) * ds2 + (long)m0 * ldd + n0;
  float* Db = D + offD;
  const int r = lane & 15;
  const int kh = (lane >> 4) << 1;  // 0 or 2
  v8f acc[NT];
#pragma unroll
  for (int j = 0; j < NT; ++j) acc[j] = v8f{0.f, 0.f, 0.f, 0.f, 0.f, 0.f, 0.f, 0.f};
  const float* ap = Ab + (long)r * lda + kh;
  const float* bp = Bb + (long)kh * ldb + r;
  for (int k0 = 0; k0 < K; k0 += 4) {
    v2f a;
    a.x = ap[0];
    a.y = ap[1];
#pragma unroll
    for (int j = 0; j < NT; ++j) {
      v2f b;
      b.x = bp[16 * j];
      b.y = bp[16 * j + ldb];
      acc[j] = __builtin_amdgcn_wmma_f32_16x16x4_f32(
          false, a, false, b, (short)0, acc[j], false, false);
    }
    ap += 4;
    bp += (long)4 * ldb;
  }
  const int half = lane >> 4;
  const float* Rb = resid ? (resid + offD) : nullptr;
#pragma unroll
  for (int j = 0; j < NT; ++j) {
#pragma unroll
    for (int i = 0; i < 8; ++i) {
      const int m = i + 8 * half;
      const long off = (long)m * ldd + 16 * j + r;
      float v = alpha * acc[j][i];
      if (flags & GF_ACCUM) v += Db[off];
      if (bias) v += bias[m0 + m];
      if (flags & GF_GELU) v = dev_gelu(v);
      if (Rb) v += Rb[off];
      Db[off] = v;
    }
  }
}

// ---------------------------------------------------------------------------
// DFT weight tables
// ---------------------------------------------------------------------------
__global__ void init_tables(float* Tcos, float* Tnsin, float* Cos2, float* Sin2,
                            float* Wc, float* Ws) {
  const int i = blockIdx.x;    // 0..127
  const int j = threadIdx.x;   // 0..127
  const float ang = (float)((i * j) & 127) * 0.0490873852123405f;  // pi/64
  const float cv = cosf(ang), sv = sinf(ang);
  Cos2[i * 128 + j] = cv;
  Sin2[i * 128 + j] = sv;
  if (j < WF_) {
    Tcos[i * WF_ + j]  = (j <= 64) ? cv : 0.f;
    Tnsin[i * WF_ + j] = (j <= 64) ? -sv : 0.f;
  }
  if (i < WF_) {
    if (i <= 64) {
      const float ck = ((i == 0) || (i == 64)) ? 1.f : 2.f;
      const float inv = 1.0f / 16384.0f;  // full 1/(H*W) of irfft2
      Wc[i * 128 + j] = ck * cv * inv;
      Ws[i * 128 + j] = -ck * sv * inv;
    } else {
      Wc[i * 128 + j] = 0.f;
      Ws[i * 128 + j] = 0.f;
    }
  }
}

// ---------------------------------------------------------------------------
// fd: dynamic low-pass filter
// ---------------------------------------------------------------------------
__global__ void mean_hw(const float* __restrict__ x, float* __restrict__ ap) {
  const long bc = blockIdx.x;
  const float* p = x + bc * HW_;
  float s = 0.f;
  for (int i = threadIdx.x; i < HW_; i += blockDim.x) s += p[i];
  __shared__ float sh[256];
  sh[threadIdx.x] = s;
  __syncthreads();
  for (int off = 128; off > 0; off >>= 1) {
    if (threadIdx.x < off) sh[threadIdx.x] += sh[threadIdx.x + off];
    __syncthreads();
  }
  if (threadIdx.x == 0) ap[bc] = sh[0] / (float)HW_;
}

__global__ void fd_lf(const float* __restrict__ ap, const float* __restrict__ fdw,
                      const float* __restrict__ g, const float* __restrict__ bb,
                      const float* __restrict__ m, const float* __restrict__ v,
                      float* __restrict__ lfraw) {
  const int t = blockIdx.x * blockDim.x + threadIdx.x;
  if (t >= B_ * 72) return;
  const int o = t % 72, b = t / 72;
  float s = 0.f;
  for (int c = 0; c < C_; ++c) s += fdw[o * C_ + c] * ap[b * C_ + c];
  s = (s - m[o]) * rsqrtf(v[o] + 1e-5f) * g[o] + bb[o];
  lfraw[t] = s;
}

__global__ void fd_softmax(const float* __restrict__ lfraw, float* __restrict__ lfw) {
  const int t = blockIdx.x * blockDim.x + threadIdx.x;
  if (t >= B_ * 8) return;
  const int gI = t % 8, b = t / 8;
  const float* src = lfraw + b * 72 + gI * 9;
  float* dst = lfw + b * 72 + gI * 9;
  float mx = src[0];
  for (int k = 1; k < 9; ++k) mx = fmaxf(mx, src[k]);
  float e[9], s = 0.f;
  for (int k = 0; k < 9; ++k) { e[k] = expf(src[k] - mx); s += e[k]; }
  for (int k = 0; k < 9; ++k) dst[k] = e[k] / s;
}

__global__ void fd_apply(const float* __restrict__ x, const float* __restrict__ lfw,
                         float* __restrict__ low, float* __restrict__ high) {
  const long i = (long)blockIdx.x * blockDim.x + threadIdx.x;
  if (i >= NCf) return;
  const int w = (int)(i % W_);
  const int h = (int)((i / W_) % H_);
  const int c = (int)((i / HW_) % C_);
  const int b = (int)(i / (HW_ * C_));
  const int g = c / 6;
  const float* wv = lfw + (b * 8 + g) * 9;
  const float* base = x + ((long)b * C_ + c) * HW_;
  float s = 0.f;
#pragma unroll
  for (int ti = 0; ti < 3; ++ti) {
    const int hr = refl(h + ti - 1, H_);
#pragma unroll
    for (int tj = 0; tj < 3; ++tj) {
      const int wr = refl(w + tj - 1, W_);
      s += wv[ti * 3 + tj] * base[(long)hr * W_ + wr];
    }
  }
  low[i] = s;
  high[i] = x[i] - s;
}

// ---------------------------------------------------------------------------
// fmgm_high: gelu(dw(3,1)(dw(1,3)(hi))) * hi   (zero-padded SAME)
// ---------------------------------------------------------------------------
__global__ void fmgm_high_k(const float* __restrict__ hi,
                            const float* __restrict__ w1, const float* __restrict__ b1,
                            const float* __restrict__ w2, const float* __restrict__ b2,
                            float* __restrict__ out) {
  const long i = (long)blockIdx.x * blockDim.x + threadIdx.x;
  if (i >= NCf) return;
  const int w = (int)(i % W_);
  const int h = (int)((i / W_) % H_);
  const int c = (int)((i / HW_) % C_);
  const float* base = hi + (i - (long)h * W_ - w);
  float acc = 0.f;
#pragma unroll
  for (int ti = 0; ti < 3; ++ti) {
    const int hh = h + ti - 1;
    if (hh < 0 || hh >= H_) continue;  // zero-pad of y1 (incl. its bias)
    float y1 = b1[c];
#pragma unroll
    for (int tj = 0; tj < 3; ++tj) {
      const int ww = w + tj - 1;
      if (ww < 0 || ww >= W_) continue;
      y1 += w1[c * 3 + tj] * base[(long)hh * W_ + ww];
    }
    acc += w2[c * 3 + ti] * y1;
  }
  acc += b2[c];
  out[i] = dev_gelu(acc) * hi[i];
}

// ---------------------------------------------------------------------------
// frequency gating: zr = Fr*gelu_r, zi = Fi*gelu_i (planes padded to 80 cols)
// ---------------------------------------------------------------------------
__global__ void freq_gate(const float* __restrict__ frq, const float* __restrict__ gb,
                          float* __restrict__ zr, float* __restrict__ zi) {
  const long i = (long)blockIdx.x * blockDim.x + threadIdx.x;
  if (i >= PLANEf) return;
  const long p = i % P80;
  const long bc = i / P80;
  const long b = bc / C_, c = bc % C_;
  const long ro = (b * 96 + c) * P80 + p;
  const long io = (b * 96 + 48 + c) * P80 + p;
  zr[i] = frq[ro] * gb[ro];
  zi[i] = frq[io] * gb[io];
}

// ---------------------------------------------------------------------------
// depthwise 3x3, zero-pad SAME, no bias
// ---------------------------------------------------------------------------
__global__ void dwconv3x3(const float* __restrict__ x, const float* __restrict__ w,
                          float* __restrict__ y, int C, long total) {
  const long i = (long)blockIdx.x * blockDim.x + threadIdx.x;
  if (i >= total) return;
  const int ww = (int)(i % W_);
  const int h = (int)((i / W_) % H_);
  const int c = (int)((i / HW_) % C);
  const float* base = x + (i - (long)h * W_ - ww);
  float s = 0.f;
#pragma unroll
  for (int ti = 0; ti < 3; ++ti) {
    const int hh = h + ti - 1;
    if (hh < 0 || hh >= H_) continue;
#pragma unroll
    for (int tj = 0; tj < 3; ++tj) {
      const int cc = ww + tj - 1;
      if (cc < 0 || cc >= W_) continue;
      s += w[c * 9 + ti * 3 + tj] * base[(long)hh * W_ + cc];
    }
  }
  y[i] = s;
}

// ---------------------------------------------------------------------------
// attention helpers
// ---------------------------------------------------------------------------
__global__ void l2norm(const float* __restrict__ x, float* __restrict__ out,
                       int nch, int cstride) {
  const int b = blockIdx.x / nch, c = blockIdx.x % nch;
  const float* p = x + ((long)b * cstride + c) * HW_;
  float s = 0.f;
  for (int i = threadIdx.x; i < HW_; i += blockDim.x) { float v = p[i]; s += v * v; }
  __shared__ float sh[256];
  sh[threadIdx.x] = s;
  __syncthreads();
  for (int off = 128; off > 0; off >>= 1) {
    if (threadIdx.x < off) sh[threadIdx.x] += sh[threadIdx.x + off];
    __syncthreads();
  }
  if (threadIdx.x == 0) out[blockIdx.x] = sqrtf(sh[0]);
}

__global__ void attn_dots(const float* __restrict__ q, const float* __restrict__ kv,
                          float* __restrict__ dots) {
  const int idx = blockIdx.x;  // (((b*4+h)*12+c)*12+d)
  const int d = idx % 12;
  const int c = (idx / 12) % 12;
  const int h = (idx / 144) % 4;
  const int b = idx / 576;
  const float* qp = q + ((long)b * 48 + h * 12 + c) * HW_;
  const float* kp = kv + ((long)b * 96 + h * 12 + d) * HW_;
  float s = 0.f;
  for (int i = threadIdx.x; i < HW_; i += blockDim.x) s += qp[i] * kp[i];
  __shared__ float sh[256];
  sh[threadIdx.x] = s;
  __syncthreads();
  for (int off = 128; off > 0; off >>= 1) {
    if (threadIdx.x < off) sh[threadIdx.x] += sh[threadIdx.x + off];
    __syncthreads();
  }
  if (threadIdx.x == 0) dots[idx] = sh[0];
}

__global__ void attn_softmax(const float* __restrict__ dots, const float* __restrict__ qn,
                             const float* __restrict__ kn, const float* __restrict__ temp,
                             float* __restrict__ attn) {
  const int t = blockIdx.x * blockDim.x + threadIdx.x;
  if (t >= B_ * 48) return;
  const int c = t % 12, h = (t / 12) % 4, b = t / 48;
  const float qv = fmaxf(qn[b * 48 + h * 12 + c], 1e-12f);
  const int base = ((b * 4 + h) * 12 + c) * 12;
  float s[12], mx = -3.0e38f;
  for (int d = 0; d < 12; ++d) {
    const float kv = fmaxf(kn[b * 48 + h * 12 + d], 1e-12f);
    s[d] = dots[base + d] / (qv * kv) * temp[h];
    mx = fmaxf(mx, s[d]);
  }
  float sum = 0.f;
  for (int d = 0; d < 12; ++d) { s[d] = expf(s[d] - mx); sum += s[d]; }
  for (int d = 0; d < 12; ++d) attn[base + d] = s[d] / sum;
}

__global__ void attn_apply(const float* __restrict__ attn, const float* __restrict__ kv,
                           float* __restrict__ out) {
  const long i = (long)blockIdx.x * blockDim.x + threadIdx.x;
  if (i >= NCf) return;
  const long p = i % HW_;
  const int ch = (int)((i / HW_) % 48);
  const int b = (int)(i / (HW_ * 48));
  const int h = ch / 12, c = ch % 12;
  const float* ar = attn + ((b * 4 + h) * 12 + c) * 12;
  const float* v = kv + ((long)b * 96 + 48 + h * 12) * HW_ + p;
  float s = 0.f;
#pragma unroll
  for (int d = 0; d < 12; ++d) s += ar[d] * v[(long)d * HW_];
  out[i] = s;
}

// ---------------------------------------------------------------------------
// host-side helpers
// ---------------------------------------------------------------------------
struct GOp { const float* p; int ld; long m, s1, s2; };

static inline void gemm(hipStream_t st, GOp A, GOp B, float* D, int ldd,
                        long dm, long ds1, long ds2, int M, int N, int K, int nb,
                        float alpha, const float* bias, const float* resid, int flags) {
  dim3 blk(32);
  if (N % 64 == 0) {
    dim3 g((unsigned)(N / 64), (unsigned)(M / 16), (unsigned)nb);
    gemm_wmma_f32<4><<<g, blk, 0, st>>>(A.p, A.ld, A.m, A.s1, A.s2,
                                        B.p, B.ld, B.m, B.s1, B.s2,
                                        D, ldd, dm, ds1, ds2, K, alpha, bias, resid, flags);
  } else {
    dim3 g((unsigned)(N / 16), (unsigned)(M / 16), (unsigned)nb);
    gemm_wmma_f32<1><<<g, blk, 0, st>>>(A.p, A.ld, A.m, A.s1, A.s2,
                                        B.p, B.ld, B.m, B.s1, B.s2,
                                        D, ldd, dm, ds1, ds2, K, alpha, bias, resid, flags);
  }
}

// conv1x1: out[b][o][p] = sum_c w[o][c] * in[b][c][p]
static inline void conv1x1(hipStream_t st, const float* w, const float* in, float* out,
                           int O, int Cin, const float* bias, const float* resid, int flags) {
  gemm(st, GOp{w, Cin, 1, 0, 0}, GOp{in, (int)HW_, BIGM, (long)Cin * HW_, 0},
       out, (int)HW_, BIGM, (long)O * HW_, 0, O, (int)HW_, Cin, B_, 1.f, bias, resid, flags);
}

static void run_fga(hipStream_t st, const float* x, const float* freq,
                    const float* qw, const float* kvw, const float* qdw,
                    const float* kvdw, const float* pw, const float* temp,
                    float* ws, float* out) {
  float* t1 = ws + o_t1;  // kv_pre (2C)
  float* t2 = ws + o_t2;  // kv (2C)
  float* t3 = ws + o_t3;  // q_pre -> outattn
  float* t4 = ws + o_t4;  // q
  float* qn = ws + o_qn;
  float* kn = ws + o_kn;
  float* dots = ws + o_dots;
  float* attn = ws + o_attn;
  const int thr = 256;
  conv1x1(st, qw, freq, t3, 48, 48, nullptr, nullptr, 0);
  dwconv3x3<<<(unsigned)((NCf + thr - 1) / thr), thr, 0, st>>>(t3, qdw, t4, 48, NCf);
  conv1x1(st, kvw, x, t1, 96, 48, nullptr, nullptr, 0);
  dwconv3x3<<<(unsigned)((2 * NCf + thr - 1) / thr), thr, 0, st>>>(t1, kvdw, t2, 96, 2 * NCf);
  l2norm<<<B_ * 48, 256, 0, st>>>(t4, qn, 48, 48);
  l2norm<<<B_ * 48, 256, 0, st>>>(t2, kn, 48, 96);
  attn_dots<<<B_ * 4 * 144, 256, 0, st>>>(t4, t2, dots);
  attn_softmax<<<2, 256, 0, st>>>(dots, qn, kn, temp, attn);
  attn_apply<<<(unsigned)((NCf + thr - 1) / thr), thr, 0, st>>>(attn, t2, t3);
  conv1x1(st, pw, t3, out, 48, 48, nullptr, nullptr, 0);
}

extern "C" void kernel_launch(void* const* d_in, const int* in_sizes, int n_in,
                              void* d_out, int out_size, void* d_ws, size_t ws_size,
                              hipStream_t stream) {
  (void)in_sizes; (void)n_in; (void)out_size; (void)ws_size;
  const float* x      = (const float*)d_in[0];
  const float* fd_w   = (const float*)d_in[1];
  const float* bn_g   = (const float*)d_in[2];
  const float* bn_b   = (const float*)d_in[3];
  const float* bn_m   = (const float*)d_in[4];
  const float* bn_v   = (const float*)d_in[5];
  const float* fh_w1  = (const float*)d_in[6];
  const float* fh_b1  = (const float*)d_in[7];
  const float* fh_w2  = (const float*)d_in[8];
  const float* fh_b2  = (const float*)d_in[9];
  const float* fl_w   = (const float*)d_in[10];
  const float* fl_b   = (const float*)d_in[11];
  const float* hfa_qw  = (const float*)d_in[12];
  const float* hfa_kvw = (const float*)d_in[13];
  const float* hfa_qdw = (const float*)d_in[14];
  const float* hfa_kvdw= (const float*)d_in[15];
  const float* hfa_pw  = (const float*)d_in[16];
  const float* hfa_t   = (const float*)d_in[17];
  const float* lfa_qw  = (const float*)d_in[18];
  const float* lfa_kvw = (const float*)d_in[19];
  const float* lfa_qdw = (const float*)d_in[20];
  const float* lfa_kvdw= (const float*)d_in[21];
  const float* lfa_pw  = (const float*)d_in[22];
  const float* lfa_t   = (const float*)d_in[23];
  const float* fp_w    = (const float*)d_in[24];
  const float* fp_b    = (const float*)d_in[25];

  float* ws = (float*)d_ws;
  float* low = ws + o_low;
  float* high = ws + o_high;
  float* hf = ws + o_hf;
  float* lfb = ws + o_lfb;
  float* ho = ws + o_ho;
  float* lo = ws + o_lo;
  float* p1 = ws + o_ho;   // reuse before fga phase
  float* p2 = ws + o_lo;
  float* p3 = ws + o_t3;
  float* p4 = ws + o_t4;
  float* frq = ws + o_t1;  // [B][96][P80]
  float* gbuf = ws + o_t2;
  float* Tcos = ws + o_Tcos;
  float* Tnsin = ws + o_Tnsin;
  float* Cos2 = ws + o_Cos2;
  float* Sin2 = ws + o_Sin2;
  float* Wc = ws + o_Wc;
  float* Wsn = ws + o_Ws;
  float* out = (float*)d_out;
  const int thr = 256;

  // tables + fd
  init_tables<<<128, 128, 0, stream>>>(Tcos, Tnsin, Cos2, Sin2, Wc, Wsn);
  mean_hw<<<B_ * C_, 256, 0, stream>>>(x, ws + o_ap);
  fd_lf<<<3, 256, 0, stream>>>(ws + o_ap, fd_w, bn_g, bn_b, bn_m, bn_v, ws + o_lfraw);
  fd_softmax<<<1, 64, 0, stream>>>(ws + o_lfraw, ws + o_lfw);
  fd_apply<<<(unsigned)(NCf / thr), thr, 0, stream>>>(x, ws + o_lfw, low, high);

  // fmgm_high
  fmgm_high_k<<<(unsigned)(NCf / thr), thr, 0, stream>>>(high, fh_w1, fh_b1, fh_w2, fh_b2, hf);

  // fmgm_low: rfft2 via DFT GEMMs -------------------------------------------
  const int Mrows = (int)(B_ * C_ * H_);  // 49152
  // stage 1 (row rDFT): R1r = low * Tcos ; R1i = low * Tnsin
  gemm(stream, GOp{low, W_, 1, 0, 0}, GOp{Tcos, WF_, 1, 0, 0},
       p1, WF_, 1, 0, 0, Mrows, WF_, W_, 1, 1.f, nullptr, nullptr, 0);
  gemm(stream, GOp{low, W_, 1, 0, 0}, GOp{Tnsin, WF_, 1, 0, 0},
       p2, WF_, 1, 0, 0, Mrows, WF_, W_, 1, 1.f, nullptr, nullptr, 0);
  // stage 2 (column DFT, batch 384): Fr = Cos2*R1r + Sin2*R1i ; Fi = Cos2*R1i - Sin2*R1r
  gemm(stream, GOp{Cos2, 128, 1, 0, 0}, GOp{p1, WF_, BIGM, P80, 0},
       frq, WF_, 48, P80, 96 * P80, H_, WF_, H_, 384, 1.f, nullptr, nullptr, 0);
  gemm(stream, GOp{Sin2, 128, 1, 0, 0}, GOp{p2, WF_, BIGM, P80, 0},
       frq, WF_, 48, P80, 96 * P80, H_, WF_, H_, 384, 1.f, nullptr, nullptr, GF_ACCUM);
  gemm(stream, GOp{Cos2, 128, 1, 0, 0}, GOp{p2, WF_, BIGM, P80, 0},
       frq + 48 * P80, WF_, 48, P80, 96 * P80, H_, WF_, H_, 384, 1.f, nullptr, nullptr, 0);
  gemm(stream, GOp{Sin2, 128, 1, 0, 0}, GOp{p1, WF_, BIGM, P80, 0},
       frq + 48 * P80, WF_, 48, P80, 96 * P80, H_, WF_, H_, 384, -1.f, nullptr, nullptr, GF_ACCUM);
  // gating: gbuf = gelu(fl_w * frq + fl_b)
  gemm(stream, GOp{fl_w, 96, 1, 0, 0}, GOp{frq, (int)P80, BIGM, 96 * P80, 0},
       gbuf, (int)P80, BIGM, 96 * P80, 0, 96, (int)P80, 96, B_, 1.f, fl_b, nullptr, GF_GELU);
  freq_gate<<<(unsigned)(PLANEf / thr), thr, 0, stream>>>(frq, gbuf, p1, p2);
  // column iDFT: Z2r = Cos2*zr - Sin2*zi ; Z2i = Cos2*zi + Sin2*zr
  gemm(stream, GOp{Cos2, 128, 1, 0, 0}, GOp{p1, WF_, BIGM, P80, 0},
       p3, WF_, BIGM, P80, 0, H_, WF_, H_, 384, 1.f, nullptr, nullptr, 0);
  gemm(stream, GOp{Sin2, 128, 1, 0, 0}, GOp{p2, WF_, BIGM, P80, 0},
       p3, WF_, BIGM, P80, 0, H_, WF_, H_, 384, -1.f, nullptr, nullptr, GF_ACCUM);
  gemm(stream, GOp{Cos2, 128, 1, 0, 0}, GOp{p2, WF_, BIGM, P80, 0},
       p4, WF_, BIGM, P80, 0, H_, WF_, H_, 384, 1.f, nullptr, nullptr, 0);
  gemm(stream, GOp{Sin2, 128, 1, 0, 0}, GOp{p1, WF_, BIGM, P80, 0},
       p4, WF_, BIGM, P80, 0, H_, WF_, H_, 384, 1.f, nullptr, nullptr, GF_ACCUM);
  // row irDFT (Hermitian): lfb = Z2r*Wc + Z2i*Ws  (1/(H*W) folded into tables)
  gemm(stream, GOp{p3, WF_, BIGM, P80, 0}, GOp{Wc, 128, 1, 0, 0},
       lfb, W_, BIGM, HW_, 0, H_, W_, WF_, 384, 1.f, nullptr, nullptr, 0);
  gemm(stream, GOp{p4, WF_, BIGM, P80, 0}, GOp{Wsn, 128, 1, 0, 0},
       lfb, W_, BIGM, HW_, 0, H_, W_, WF_, 384, 1.f, nullptr, nullptr, GF_ACCUM);

  // two frequency-guided attentions (planes p1..p4 are dead from here)
  run_fga(stream, x, hf, hfa_qw, hfa_kvw, hfa_qdw, hfa_kvdw, hfa_pw, hfa_t, ws, ho);
  run_fga(stream, x, lfb, lfa_qw, lfa_kvw, lfa_qdw, lfa_kvdw, lfa_pw, lfa_t, ws, lo);

  // out = fp_w[:, :48]*ho + fp_w[:, 48:]*lo + fp_b + x
  gemm(stream, GOp{fp_w, 96, 1, 0, 0}, GOp{ho, (int)HW_, BIGM, 48 * HW_, 0},
       out, (int)HW_, BIGM, 48 * HW_, 0, 48, (int)HW_, 48, B_, 1.f, nullptr, nullptr, 0);
  gemm(stream, GOp{fp_w + 48, 96, 1, 0, 0}, GOp{lo, (int)HW_, BIGM, 48 * HW_, 0},
       out, (int)HW_, BIGM, 48 * HW_, 0, 48, (int)HW_, 48, B_, 1.f, fp_b, x, GF_ACCUM);
}